// NURBSSurface_41772851921172
// MI455X (gfx1250) — compile-verified
//
#include <hip/hip_runtime.h>
#include <math.h>

typedef __attribute__((ext_vector_type(2))) float v2f;
typedef __attribute__((ext_vector_type(8))) float v8f;

#define DEG   3
#define NCP   32
#define OUT   1024
#define EPSF  1e-5f

// workspace layout (in floats)
#define OFF_AX   0
#define OFF_DAX  (OFF_AX  + OUT*NCP)
#define OFF_AY   (OFF_DAX + OUT*NCP)
#define OFF_DAY  (OFF_AY  + OUT*NCP)
#define OFF_M1   (OFF_DAY + OUT*NCP)      // [3][OUT][NCP]
#define OFF_DM1  (OFF_M1  + 3*OUT*NCP)    // [3][OUT][NCP]
#define M1_STR   (OUT*NCP)

// ---------------------------------------------------------------------------
// Kernel 1: Cox-de Boor basis + first derivative, scattered into dense
// banded rows Ax/dAx (x direction) and Ay/dAy (y direction).
// ---------------------------------------------------------------------------
__global__ void nurbs_basis_kernel(const float* __restrict__ kx,
                                   const float* __restrict__ ky,
                                   float* __restrict__ w) {
    int tid = blockIdx.x * blockDim.x + threadIdx.x;
    if (tid >= 2 * OUT) return;
    bool isY = tid >= OUT;
    int i = isY ? (tid - OUT) : tid;
    const float* knots = isY ? ky : kx;
    float* A  = w + (isY ? OFF_AY  : OFF_AX);
    float* dA = w + (isY ? OFF_DAY : OFF_DAX);

    float t = EPSF + (1.0f - 2.0f * EPSF) * (float)i / (float)(OUT - 1);

    // span = clip(searchsorted(knots, t, 'right') - 1, DEG, NCP-1)
    int span = DEG;
    for (int k = DEG + 1; k <= NCP - 1; ++k)
        if (knots[k] <= t) span = k;

    // ndu table: ndu[j][r] (j>=r diag) knot diffs, ndu[r][j] basis values
    float ndu[DEG + 1][DEG + 1];
    float left[DEG + 1], right[DEG + 1];
    ndu[0][0] = 1.0f;
    for (int j = 1; j <= DEG; ++j) {
        left[j]  = t - knots[span + 1 - j];
        right[j] = knots[span + j] - t;
        float saved = 0.0f;
        for (int r = 0; r < j; ++r) {
            ndu[j][r] = right[r + 1] + left[j - r];
            float tmp = ndu[r][j - 1] / ndu[j][r];
            ndu[r][j] = saved + right[r + 1] * tmp;
            saved = left[j - r] * tmp;
        }
        ndu[j][j] = saved;
    }
    float d0[DEG + 1], d1[DEG + 1];
    for (int j = 0; j <= DEG; ++j) d0[j] = ndu[j][DEG];
    // A2.3 specialized to first derivative (n=1)
    for (int r = 0; r <= DEG; ++r) {
        float d = 0.0f;
        if (r >= 1)       d += (1.0f / ndu[DEG][r - 1]) * ndu[r - 1][DEG - 1];
        if (r <= DEG - 1) d -= (1.0f / ndu[DEG][r])     * ndu[r][DEG - 1];
        d1[r] = d * (float)DEG;
    }

    float* rowA  = A  + i * NCP;
    float* rowdA = dA + i * NCP;
    #pragma unroll
    for (int c = 0; c < NCP; ++c) { rowA[c] = 0.0f; rowdA[c] = 0.0f; }
    int c0 = span - DEG;
    #pragma unroll
    for (int j = 0; j <= DEG; ++j) { rowA[c0 + j] = d0[j]; rowdA[c0 + j] = d1[j]; }
}

// ---------------------------------------------------------------------------
// Kernel 2: stage-1 GEMMs   M1_d = Ax * CP_d,  dM1_d = dAx * CP_d
// grid = (64 row tiles, 6 variants = {Ax,dAx} x {d=0,1,2}), 1 wave per block.
// ---------------------------------------------------------------------------
__global__ void nurbs_stage1_kernel(const float* __restrict__ cp,
                                    float* __restrict__ w) {
    int lane = threadIdx.x;                 // 0..31
    int rt   = blockIdx.x;                  // row tile 0..63
    int var  = blockIdx.y;                  // 0..5
    int d    = var % 3;
    const float* A = w + (var < 3 ? OFF_AX : OFF_DAX);
    float* Mout    = w + (var < 3 ? OFF_M1 : OFF_DM1) + d * M1_STR;

    int m   = lane & 15;
    int kk  = (lane < 16) ? 0 : 2;
    int row = rt * 16 + m;

    v8f acc0 = {};
    v8f acc1 = {};
    for (int kc = 0; kc < 8; ++kc) {
        int k0 = kc * 4;
        v2f a;
        a.x = A[row * NCP + k0 + kk];
        a.y = A[row * NCP + k0 + kk + 1];
        // B[k][n] = CP[k][n][d]   (column block n0 = 0 and n0 = 16)
        v2f b0, b1;
        b0.x = cp[(k0 + kk)     * (NCP * 3) + m * 3 + d];
        b0.y = cp[(k0 + kk + 1) * (NCP * 3) + m * 3 + d];
        b1.x = cp[(k0 + kk)     * (NCP * 3) + (16 + m) * 3 + d];
        b1.y = cp[(k0 + kk + 1) * (NCP * 3) + (16 + m) * 3 + d];
        acc0 = __builtin_amdgcn_wmma_f32_16x16x4_f32(false, a, false, b0,
                                                     (short)0, acc0, false, false);
        acc1 = __builtin_amdgcn_wmma_f32_16x16x4_f32(false, a, false, b1,
                                                     (short)0, acc1, false, false);
    }
    #pragma unroll
    for (int e = 0; e < 8; ++e) {
        int mm = e + ((lane >> 4) << 3);
        int n  = lane & 15;
        Mout[(rt * 16 + mm) * NCP + n]      = acc0[e];
        Mout[(rt * 16 + mm) * NCP + 16 + n] = acc1[e];
    }
}

// ---------------------------------------------------------------------------
// Kernel 3: stage-2 GEMMs + epilogue. One wave per 16x16 output tile.
//   S_d = M1_d * Ay^T ; Sy_d = M1_d * dAy^T ; Sx_d = dM1_d * Ay^T
//   point = (S0,S1,S2,0); normal = normalize(cross(Sy, Sx))
// ---------------------------------------------------------------------------
__global__ void nurbs_stage2_kernel(float* __restrict__ w,
                                    float* __restrict__ out) {
    __shared__ __align__(16) float sM1[3][16][NCP];
    __shared__ __align__(16) float sDM1[3][16][NCP];
    __shared__ __align__(16) float sAy[16][NCP];
    __shared__ __align__(16) float sDAy[16][NCP];

    int lane = threadIdx.x;
    int px0  = blockIdx.x * 16;
    int py0  = blockIdx.y * 16;

    for (int i = lane; i < 3 * 16 * NCP; i += 32) {
        int d = i / (16 * NCP);
        int r = (i / NCP) & 15;
        int c = i % NCP;
        sM1[d][r][c]  = w[OFF_M1  + d * M1_STR + (px0 + r) * NCP + c];
        sDM1[d][r][c] = w[OFF_DM1 + d * M1_STR + (px0 + r) * NCP + c];
    }
    for (int i = lane; i < 16 * NCP; i += 32) {
        int r = i / NCP, c = i % NCP;
        sAy[r][c]  = w[OFF_AY  + (py0 + r) * NCP + c];
        sDAy[r][c] = w[OFF_DAY + (py0 + r) * NCP + c];
    }
    __syncthreads();

    int m  = lane & 15;
    int kk = (lane < 16) ? 0 : 2;

    v8f S[3]  = {{}, {}, {}};
    v8f Sx[3] = {{}, {}, {}};
    v8f Sy[3] = {{}, {}, {}};

    for (int kc = 0; kc < 8; ++kc) {
        int k0 = kc * 4 + kk;
        // B[k][n] = Ay[py0+n][k] : mirrored fragment, lane n = m
        v2f bA = *(const v2f*)&sAy[m][k0];
        v2f bD = *(const v2f*)&sDAy[m][k0];
        #pragma unroll
        for (int d = 0; d < 3; ++d) {
            v2f aM = *(const v2f*)&sM1[d][m][k0];
            v2f aD = *(const v2f*)&sDM1[d][m][k0];
            S[d]  = __builtin_amdgcn_wmma_f32_16x16x4_f32(false, aM, false, bA,
                                                          (short)0, S[d],  false, false);
            Sy[d] = __builtin_amdgcn_wmma_f32_16x16x4_f32(false, aM, false, bD,
                                                          (short)0, Sy[d], false, false);
            Sx[d] = __builtin_amdgcn_wmma_f32_16x16x4_f32(false, aD, false, bA,
                                                          (short)0, Sx[d], false, false);
        }
    }

    float* out_pt  = out;
    float* out_nrm = out + 4 * OUT * OUT;
    #pragma unroll
    for (int e = 0; e < 8; ++e) {
        int mm = e + ((lane >> 4) << 3);
        int n  = lane & 15;
        int p  = (px0 + mm) * OUT + (py0 + n);

        float4 pt = make_float4(S[0][e], S[1][e], S[2][e], 0.0f);
        *(float4*)&out_pt[(size_t)p * 4] = pt;

        float ax = Sy[0][e], ay = Sy[1][e], az = Sy[2][e];
        float bx = Sx[0][e], by = Sx[1][e], bz = Sx[2][e];
        float nx = ay * bz - az * by;
        float ny = az * bx - ax * bz;
        float nz = ax * by - ay * bx;
        float len = sqrtf(nx * nx + ny * ny + nz * nz);
        float inv = 1.0f / fmaxf(len, 1e-12f);
        out_nrm[(size_t)p * 3 + 0] = nx * inv;
        out_nrm[(size_t)p * 3 + 1] = ny * inv;
        out_nrm[(size_t)p * 3 + 2] = nz * inv;
    }
}

extern "C" void kernel_launch(void* const* d_in, const int* in_sizes, int n_in,
                              void* d_out, int out_size, void* d_ws, size_t ws_size,
                              hipStream_t stream) {
    const float* cp = (const float*)d_in[0];   // (32,32,3)
    const float* kx = (const float*)d_in[1];   // (1,36)
    const float* ky = (const float*)d_in[2];   // (1,36)
    float* w   = (float*)d_ws;
    float* out = (float*)d_out;
    (void)in_sizes; (void)n_in; (void)out_size; (void)ws_size;

    hipLaunchKernelGGL(nurbs_basis_kernel,  dim3(16),     dim3(128), 0, stream, kx, ky, w);
    hipLaunchKernelGGL(nurbs_stage1_kernel, dim3(64, 6),  dim3(32),  0, stream, cp, w);
    hipLaunchKernelGGL(nurbs_stage2_kernel, dim3(64, 64), dim3(32),  0, stream, w, out);
}